// autoencoder_85444079386734
// MI455X (gfx1250) — compile-verified
//
#include <hip/hip_runtime.h>
#include <math.h>

// ---------------------------------------------------------------------------
// Tiny autoencoder forward on gfx1250 (MI455X), single wave32, fp32 WMMA.
//
// Every dense layer is y = [W | b] * [h ; 1] via V_WMMA_F32_16X16X4_F32 tiles.
// No predication anywhere in the hot path: out-of-range rows on the second
// M-tile are *clamped duplicates* of the last valid row; their (finite)
// garbage results land in LDS padding slots 24..31 that no layer reads.
//
// d_out layout (168 floats): out[24] | in_diff_sum[64] | lat_diff_sum[64] |
//                            lat_repr[16]
// ---------------------------------------------------------------------------

typedef __attribute__((ext_vector_type(2))) float v2f;
typedef __attribute__((ext_vector_type(8))) float v8f;

#define ACT_NONE 0
#define ACT_RELU 1
#define ACT_TANH 2

// One dense layer  hout[0..O) = act(W[OxN] * hin[0..N) + b), O<=32, N%4==0.
//   A tile : 16 rows of W, K=4 slab. lane = M%16 ; VGPR v, half-wave h:
//            K = kbase + 2*h + v
//   B tile : activation chunk broadcast across all 16 columns; same K map.
//   C/D    : VGPR r, lane L -> M = r + 8*(L>=16), N = L%16 ; all columns
//            identical, so every lane stores its 8 rows (duplicates benign).
template <int O, int N, int ACT>
__device__ __forceinline__ void dense_layer_wmma(
    const float* __restrict__ W, const float* __restrict__ b,
    const float* __restrict__ hin, float* __restrict__ hout,
    int lane, float kmask /* 1.0 for lanes 0..15, 0.0 for 16..31 */)
{
    constexpr int NTILES  = (O + 15) >> 4;
    constexpr int KCHUNKS = N >> 2;

    const int mrow = lane & 15;             // row within M-tile
    const int half = lane >> 4;             // 0 or 1
    const int kpar = half * 2;              // K sub-offset per half-wave

#pragma unroll
    for (int t = 0; t < NTILES; ++t) {
        const int m  = t * 16 + mrow;
        const int mc = (m < O) ? m : (O - 1);   // clamp: rows >= O duplicate
        const float* __restrict__ Wr = W + mc * N;
        v8f acc = {};
#pragma unroll
        for (int c = 0; c < KCHUNKS; ++c) {
            const int k0 = c * 4 + kpar;
            v2f a, bb;
            a.x  = Wr[k0];                  // unconditional -> b64 clauses
            a.y  = Wr[k0 + 1];
            bb.x = hin[k0];                 // activation broadcast
            bb.y = hin[k0 + 1];
            acc = __builtin_amdgcn_wmma_f32_16x16x4_f32(
                false, a, false, bb, (short)0, acc, false, false);
        }
        {   // bias chunk: A[m][N] = b[m], B[N][n] = 1  (K slots N+1..N+3 zero)
            v2f a, bb;
            a.x  = b[mc] * kmask;           // unconditional load, masked by mul
            a.y  = 0.0f;
            bb.x = kmask;
            bb.y = 0.0f;
            acc = __builtin_amdgcn_wmma_f32_16x16x4_f32(
                false, a, false, bb, (short)0, acc, false, false);
        }
        // Branch-free writeback: lane L stores rows t*16 + 8*half + r.
        // t==1 rows 24..31 hold finite duplicate-row garbage -> padding slots.
        const int base = t * 16 + half * 8;
#pragma unroll
        for (int r = 0; r < 8; ++r) {
            float v = acc[r];
            if (ACT == ACT_RELU)      v = fmaxf(v, 0.0f);
            else if (ACT == ACT_TANH) v = tanhf(v);
            hout[base + r] = v;
        }
    }
    __syncthreads();   // single-wave WG: orders LDS for the next layer
}

__global__ __launch_bounds__(32) void autoencoder_85444079386734_kernel(
    const float* __restrict__ x,
    const float* __restrict__ ew0, const float* __restrict__ eb0,
    const float* __restrict__ ew1, const float* __restrict__ eb1,
    const float* __restrict__ ew2, const float* __restrict__ eb2,
    const float* __restrict__ ew3, const float* __restrict__ eb3,
    const float* __restrict__ dw0, const float* __restrict__ db0,
    const float* __restrict__ dw1, const float* __restrict__ db1,
    const float* __restrict__ dw2, const float* __restrict__ db2,
    const float* __restrict__ dw3, const float* __restrict__ db3,
    float* __restrict__ out)
{
    __shared__ float buf0[32];
    __shared__ float buf1[32];
    __shared__ float ybuf[32];

    const int lane = threadIdx.x;   // 32 threads == 1 wave32
    const float kmask = (lane < 16) ? 1.0f : 0.0f;

    // stage input vector into LDS (branch-free: lanes 24..31 rewrite slot 23)
    {
        const int i = lane < 24 ? lane : 23;
        buf0[i] = x[i];
    }
    __syncthreads();

    // ---- encoder: 3x (Linear 24x24 + ReLU), then Linear 16x24 -> latent y
    dense_layer_wmma<24, 24, ACT_RELU>(ew0, eb0, buf0, buf1, lane, kmask);
    dense_layer_wmma<24, 24, ACT_RELU>(ew1, eb1, buf1, buf0, lane, kmask);
    dense_layer_wmma<24, 24, ACT_RELU>(ew2, eb2, buf0, buf1, lane, kmask);
    dense_layer_wmma<16, 24, ACT_NONE>(ew3, eb3, buf1, ybuf, lane, kmask);

    // ---- decoder: Linear 24x16 + ReLU, 2x (Linear 24x24 + ReLU),
    //               Linear 24x24 + Tanh
    dense_layer_wmma<24, 16, ACT_RELU>(dw0, db0, ybuf, buf0, lane, kmask);
    dense_layer_wmma<24, 24, ACT_RELU>(dw1, db1, buf0, buf1, lane, kmask);
    dense_layer_wmma<24, 24, ACT_RELU>(dw2, db2, buf1, buf0, lane, kmask);
    dense_layer_wmma<24, 24, ACT_TANH>(dw3, db3, buf0, buf1, lane, kmask);

    // ---- outputs -----------------------------------------------------------
    // out[24]: branch-free, lanes 24..31 duplicate-store slot 23 (same value)
    {
        const int i = lane < 24 ? lane : 23;
        out[i] = buf1[i];
    }

    // in_diff_sum[64]: pairwise L2 over rows of x.reshape(8,3); 2 entries/lane.
    // lat_diff_sum[64]: faithful bug reproduction -> all zeros.
#pragma unroll
    for (int p = 0; p < 2; ++p) {
        const int e = lane + 32 * p;
        const int i = e >> 3;
        const int j = e & 7;
        const float d0 = x[3 * i + 0] - x[3 * j + 0];
        const float d1 = x[3 * i + 1] - x[3 * j + 1];
        const float d2 = x[3 * i + 2] - x[3 * j + 2];
        out[24 + e] = sqrtf(d0 * d0 + d1 * d1 + d2 * d2);
        out[88 + e] = 0.0f;
    }

    // lat_repr[16] = latent y (lanes 16..31 duplicate-store slot 15)
    {
        const int i = lane < 16 ? lane : 15;
        out[152 + i] = ybuf[i];
    }
}

extern "C" void kernel_launch(void* const* d_in, const int* in_sizes, int n_in,
                              void* d_out, int out_size, void* d_ws, size_t ws_size,
                              hipStream_t stream)
{
    (void)in_sizes; (void)n_in; (void)d_ws; (void)ws_size; (void)out_size;
    const float* x   = (const float*)d_in[0];
    const float* ew0 = (const float*)d_in[1];
    const float* eb0 = (const float*)d_in[2];
    const float* ew1 = (const float*)d_in[3];
    const float* eb1 = (const float*)d_in[4];
    const float* ew2 = (const float*)d_in[5];
    const float* eb2 = (const float*)d_in[6];
    const float* ew3 = (const float*)d_in[7];
    const float* eb3 = (const float*)d_in[8];
    const float* dw0 = (const float*)d_in[9];
    const float* db0 = (const float*)d_in[10];
    const float* dw1 = (const float*)d_in[11];
    const float* db1 = (const float*)d_in[12];
    const float* dw2 = (const float*)d_in[13];
    const float* db2 = (const float*)d_in[14];
    const float* dw3 = (const float*)d_in[15];
    const float* db3 = (const float*)d_in[16];

    autoencoder_85444079386734_kernel<<<1, 32, 0, stream>>>(
        x, ew0, eb0, ew1, eb1, ew2, eb2, ew3, eb3,
        dw0, db0, dw1, db1, dw2, db2, dw3, db3,
        (float*)d_out);
}